// EnsembleLinear_80161269612822
// MI455X (gfx1250) — compile-verified
//
#include <hip/hip_runtime.h>

// EnsembleLinear on MI455X (gfx1250): 32 independent fp32 GEMMs
//   per detector d: out[b, o] = sum_k x[b, d*256+k] * W[d*256+o, d*256+k]
//   M=4096 (batch), N=256, K=256, 32 detectors.
// fp32 matrix path: V_WMMA_F32_16X16X4_F32, LDS double-buffered K-slabs.
// Each wave computes a 64x64 register tile (16 accumulators) so every
// k-step issues 16 WMMAs against 8 LDS b64 fragment reads (4:1 after
// ds_load_2addr fusion), keeping the fp32 WMMA pipe covered.

typedef float v2f __attribute__((ext_vector_type(2)));
typedef float v4f __attribute__((ext_vector_type(4)));
typedef float v8f __attribute__((ext_vector_type(8)));

#define DIN        8192      // row stride of x, W, out (floats)
#define KT         16        // K slab per LDS stage
#define NSTAGES    16        // 256 / KT
#define LDSS       20        // LDS row stride (floats): 20*m mod 64 distinct for m<16,
                             // keeps b64 reads bank-conflict-free and b128 stores 16B aligned

__global__ __launch_bounds__(256)
void ensemble_linear_wmma_f32(const float* __restrict__ x,
                              const float* __restrict__ W,
                              float* __restrict__ out)
{
    // A slab: 128 batch rows x 16 k ; B slab: 256 output cols x 16 k (W rows, transpose-free)
    __shared__ float sA[2][128 * LDSS];   // 10 KB per buffer
    __shared__ float sB[2][256 * LDSS];   // 20 KB per buffer  (total 60 KB)

    const int blk = blockIdx.x;          // 1024 blocks
    const int d   = blk >> 5;            // detector 0..31
    const int mt  = blk & 31;            // 128-row batch tile 0..31
    const int m0  = mt * 128;

    const int tid  = threadIdx.x;
    const int lane = tid & 31;
    const int wave = tid >> 5;           // 8 waves: 2 (M) x 4 (N)
    const int wm   = wave >> 2;          // 0..1  -> 64-row strip
    const int wn   = wave & 3;           // 0..3  -> 64-col strip
    const int lm   = lane & 15;          // row/col within 16
    const int kh   = (lane >> 4) << 1;   // K half offset: 0 or 2 (fp32 A/B layout)
    const int mh   = (lane >> 4) << 3;   // D row half offset: 0 or 8

    // global tile bases (diagonal W block of detector d; WG covers all 256 outputs)
    const float* xg = x + (size_t)m0 * DIN + d * 256;
    const float* wg = W + (size_t)(d * 256) * DIN + d * 256;

    // cooperative slab load mapping:
    //   A: 128x16 = 512 float4 = 256 thr x 2 ;  B: 256x16 = 1024 float4 = 256 thr x 4
    const int r0 = tid >> 2;             // base row, +64 per iteration
    const int cg = (tid & 3) << 2;       // col group *4 within the 16-wide slab

    v4f ra[2], rb[4];

    // ---- prologue: stage 0 into buffer 0 ----
    {
#pragma unroll
        for (int j = 0; j < 2; ++j)
            ra[j] = *(const v4f*)(xg + (size_t)(r0 + j * 64) * DIN + cg);
#pragma unroll
        for (int j = 0; j < 4; ++j)
            rb[j] = *(const v4f*)(wg + (size_t)(r0 + j * 64) * DIN + cg);
#pragma unroll
        for (int j = 0; j < 2; ++j)
            *(v4f*)&sA[0][(r0 + j * 64) * LDSS + cg] = ra[j];
#pragma unroll
        for (int j = 0; j < 4; ++j)
            *(v4f*)&sB[0][(r0 + j * 64) * LDSS + cg] = rb[j];
    }
    __syncthreads();

    v8f acc[4][4];
#pragma unroll
    for (int mi = 0; mi < 4; ++mi)
#pragma unroll
        for (int ni = 0; ni < 4; ++ni)
            acc[mi][ni] = 0.0f;

    const int rowA = (wm * 64 + lm) * LDSS;   // + mi*16*LDSS per m-tile
    const int rowB = (wn * 64 + lm) * LDSS;   // + ni*16*LDSS per n-tile

    for (int s = 0; s < NSTAGES; ++s) {
        const int cur = s & 1;
        const int nxt = cur ^ 1;

        // issue next slab's global loads early (hide HBM latency behind WMMA)
        if (s + 1 < NSTAGES) {
            const int k0 = (s + 1) * KT;
#pragma unroll
            for (int j = 0; j < 2; ++j)
                ra[j] = *(const v4f*)(xg + (size_t)(r0 + j * 64) * DIN + k0 + cg);
#pragma unroll
            for (int j = 0; j < 4; ++j)
                rb[j] = *(const v4f*)(wg + (size_t)(r0 + j * 64) * DIN + k0 + cg);
        }

        const float* A  = sA[cur];
        const float* Bt = sB[cur];
#pragma unroll
        for (int kk = 0; kk < KT; kk += 4) {
            // fp32 A layout: lane holds A[lm][kh + v], v=0,1 -> contiguous b64 pair
            v2f af[4], bf[4];
#pragma unroll
            for (int mi = 0; mi < 4; ++mi)
                af[mi] = *(const v2f*)&A[rowA + mi * 16 * LDSS + kk + kh];
#pragma unroll
            for (int ni = 0; ni < 4; ++ni)
                bf[ni] = *(const v2f*)&Bt[rowB + ni * 16 * LDSS + kk + kh];
#pragma unroll
            for (int mi = 0; mi < 4; ++mi)
#pragma unroll
                for (int ni = 0; ni < 4; ++ni)
                    acc[mi][ni] = __builtin_amdgcn_wmma_f32_16x16x4_f32(
                        false, af[mi], false, bf[ni], (short)0, acc[mi][ni],
                        false, false);
        }

        // write next slab; prior-iteration barrier guarantees nxt is no longer read
        if (s + 1 < NSTAGES) {
#pragma unroll
            for (int j = 0; j < 2; ++j)
                *(v4f*)&sA[nxt][(r0 + j * 64) * LDSS + cg] = ra[j];
#pragma unroll
            for (int j = 0; j < 4; ++j)
                *(v4f*)&sB[nxt][(r0 + j * 64) * LDSS + cg] = rb[j];
        }
        __syncthreads();
    }

    // ---- epilogue: D layout -> global (VGPR r: row = r + mh, col = lm) ----
    float* og = out + (size_t)(m0 + wm * 64) * DIN + d * 256 + wn * 64;
#pragma unroll
    for (int mi = 0; mi < 4; ++mi)
#pragma unroll
        for (int ni = 0; ni < 4; ++ni)
#pragma unroll
            for (int r = 0; r < 8; ++r)
                og[(size_t)(mi * 16 + mh + r) * DIN + ni * 16 + lm] = acc[mi][ni][r];
}

extern "C" void kernel_launch(void* const* d_in, const int* in_sizes, int n_in,
                              void* d_out, int out_size, void* d_ws, size_t ws_size,
                              hipStream_t stream) {
    (void)in_sizes; (void)n_in; (void)out_size; (void)d_ws; (void)ws_size;
    const float* x = (const float*)d_in[0];   // [4096, 8192] fp32
    const float* W = (const float*)d_in[1];   // [8192, 8192] fp32
    float* out     = (float*)d_out;           // [4096, 8192] fp32

    dim3 grid(32 * 32);       // detectors * M-tiles = 1024 WGs (each WG: 128x256 tile)
    dim3 block(256);          // 8 waves of 32
    hipLaunchKernelGGL(ensemble_linear_wmma_f32, grid, block, 0, stream, x, W, out);
}